// DecoderRNN_73967926772361
// MI455X (gfx1250) — compile-verified
//
#include <hip/hip_runtime.h>

// ---------------------------------------------------------------------------
// DecoderRNN (attention + 4-layer GRU + BN/sigmoid scoring head) for gfx1250.
// All GEMMs use V_WMMA_F32_16X16X4_F32 (native fp32 WMMA, wave32).
// Recurrence is serialized via kernel boundaries (deterministic, no atomics).
// Working set (~130MB weights+keys+values) is L2-resident on MI455X (192MB).
// Streaming operands get global_prefetch_b8 (speculative, OOB-safe) because
// the chain-limited kernels run at low occupancy and can't hide L2 latency
// with wave switching alone.
// ---------------------------------------------------------------------------

#define NUM_CHARS 34
#define KEY_SIZE  128
#define VAL_SIZE  256
#define HID       512
#define T_ENC     1024
#define BATCH     64
#define SEQ_L     250
#define G0        (HID + VAL_SIZE)   // 768, cell0 hidden width
#define BN_EPS    1e-5f

typedef float v2f __attribute__((ext_vector_type(2)));
typedef float v4f __attribute__((ext_vector_type(4)));
typedef float v8f __attribute__((ext_vector_type(8)));

// D = A(16x4) * B(4x16) + C, fp32. 8 args: (neg_a, A, neg_b, B, c_mod, C, reuse_a, reuse_b)
__device__ __forceinline__ v8f wmma_k4(v2f a, v2f b, v8f c) {
  return __builtin_amdgcn_wmma_f32_16x16x4_f32(false, a, false, b, (short)0, c, false, false);
}

__device__ __forceinline__ float sigmoidf_(float x) { return 1.0f / (1.0f + __expf(-x)); }
// tanh(x) = 1 - 2/(exp(2x)+1)  (stable at both tails, uses v_exp_f32)
__device__ __forceinline__ float tanhf_(float x) { return 1.0f - 2.0f / (__expf(2.0f * x) + 1.0f); }

// ---------------------------------------------------------------------------
// Init: broadcast learned initial hiddens; write constant first_hidden half of
// the cell0 recurrent input [ctx | fh] (ctx half is rewritten every step).
// ---------------------------------------------------------------------------
__global__ void init_kernel(const float* __restrict__ fh,
                            const float* __restrict__ i1,
                            const float* __restrict__ i2,
                            const float* __restrict__ i3,
                            float* __restrict__ h0init,
                            float* __restrict__ h1,
                            float* __restrict__ h2,
                            float* __restrict__ h3) {
  int i = blockIdx.x * 256 + threadIdx.x;      // 0 .. 64*512-1
  int b = i >> 9, d = i & (HID - 1);
  h1[i] = i1[d];
  h2[i] = i2[d];
  h3[i] = i3[d];
  h0init[(size_t)b * G0 + VAL_SIZE + d] = fh[d];
}

// ---------------------------------------------------------------------------
// Attention: one block per batch row. q = h3 @ wq^T + bq; masked softmax over
// T_ENC energies; ctx = att @ values  -> written into h0init[:, 0:256].
// (softmax*mask renormalized == masked softmax, algebraically identical)
// ---------------------------------------------------------------------------
__global__ void attn_kernel(const float* __restrict__ h3,
                            const float* __restrict__ wq,
                            const float* __restrict__ bq,
                            const float* __restrict__ keys,     // [T,B,K]
                            const float* __restrict__ values,   // [T,B,V]
                            const int*   __restrict__ masks,
                            float* __restrict__ h0init) {
  const int b = blockIdx.x;
  const int tid = threadIdx.x;                 // 0..255
  __shared__ float sq[KEY_SIZE];
  __shared__ float satt[T_ENC];
  __shared__ float red[256];

  // ---- q: 2 threads per output k, 256-wide partial dots ----
  {
    int k = tid >> 1, half = tid & 1;
    const float* wrow = wq + (size_t)k * HID + half * 256;
    const float* hrow = h3 + (size_t)b * HID + half * 256;
    float s = 0.f;
    #pragma unroll 4
    for (int d = 0; d < 256; d += 4) {
      v4f w = *(const v4f*)(wrow + d);
      v4f h = *(const v4f*)(hrow + d);
      s += w.x * h.x + w.y * h.y + w.z * h.z + w.w * h.w;
    }
    red[tid] = s;
  }
  __syncthreads();
  if (tid < KEY_SIZE) sq[tid] = red[2 * tid] + red[2 * tid + 1] + bq[tid];
  __syncthreads();

  // ---- energies (4 per thread) + block max ----
  float e[4];
  float mx = -3.4e38f;
  #pragma unroll
  for (int i = 0; i < 4; ++i) {
    int tt = tid + i * 256;
    const float* krow = keys + ((size_t)tt * BATCH + b) * KEY_SIZE;
    if (i < 3) {  // prefetch next strided key row (speculative, RT)
      __builtin_prefetch(keys + ((size_t)(tt + 256) * BATCH + b) * KEY_SIZE, 0, 3);
    }
    float s = 0.f;
    #pragma unroll 4
    for (int k = 0; k < KEY_SIZE; k += 4) {
      v4f kv = *(const v4f*)(krow + k);
      s += sq[k] * kv.x + sq[k + 1] * kv.y + sq[k + 2] * kv.z + sq[k + 3] * kv.w;
    }
    e[i] = s;
    mx = fmaxf(mx, s);
  }
  red[tid] = mx;
  __syncthreads();
  for (int off = 128; off > 0; off >>= 1) {
    if (tid < off) red[tid] = fmaxf(red[tid], red[tid + off]);
    __syncthreads();
  }
  float M = red[0];
  __syncthreads();

  // ---- masked exp + block sum ----
  int mb = masks[b];
  float s = 0.f;
  #pragma unroll
  for (int i = 0; i < 4; ++i) {
    int tt = tid + i * 256;
    float v = (tt < mb) ? __expf(e[i] - M) : 0.f;
    satt[tt] = v;
    s += v;
  }
  red[tid] = s;
  __syncthreads();
  for (int off = 128; off > 0; off >>= 1) {
    if (tid < off) red[tid] += red[tid + off];
    __syncthreads();
  }
  float scale = 1.f / fmaxf(red[0], 1e-12f);

  // ---- ctx[v] = sum_t att[t] * values[t,b,v]; thread = one v column ----
  float acc = 0.f;
  const float* vcol = values + (size_t)b * VAL_SIZE + tid;
  for (int tt = 0; tt < T_ENC; tt += 8) {
    // prefetch 24 rows ahead (speculative; OOB prefetches are dropped)
    __builtin_prefetch(vcol + (size_t)(tt + 24) * (BATCH * VAL_SIZE), 0, 3);
    #pragma unroll
    for (int u = 0; u < 8; ++u) {
      acc += satt[tt + u] * vcol[(size_t)(tt + u) * (BATCH * VAL_SIZE)];
    }
  }
  h0init[(size_t)b * G0 + tid] = acc * scale;
}

// ---------------------------------------------------------------------------
// GRU cell: h' = (1-z)*n + z*h  with  r,z from sigmoid, n from tanh.
// gi = X @ Wih^T (+bih), gh = Hprev @ Whh^T (+bhh).  Block = 4 waves covering
// M=64 rows (4 x 16 tiles), 16 output columns per block.  Each wave keeps six
// v8f WMMA accumulators (r/z/n x ih/hh) and fuses the gate math in registers.
// ONE_HOT: x is one-hot(char) -> gi is a column gather of Wih (no GEMM).
// WMMA fp32 layouts (ISA 7.12.2): A/B lane = index&15 selects row(M)/col(N),
// lane>=16 selects K pair {2,3}; D vgpr e -> row e (+8 for lanes>=16).
// K-loops are chunked by 64 floats with one prefetch per stream per chunk.
// ---------------------------------------------------------------------------
template <int IN_DIM, int H_DIM, bool ONE_HOT>
__global__ void gru_kernel(const float* __restrict__ X,      // [64, IN_DIM] (unused if ONE_HOT)
                           const float* __restrict__ Hprev,  // [64, H_DIM]
                           const float* __restrict__ Wih,    // [3*H_DIM, IN_DIM]
                           const float* __restrict__ Whh,    // [3*H_DIM, H_DIM]
                           const float* __restrict__ bih,
                           const float* __restrict__ bhh,
                           const int*   __restrict__ seq,    // [B, L] (ONE_HOT only)
                           int t,
                           float* __restrict__ Hout,         // [64, H_DIM]
                           float* __restrict__ outs) {       // optional [B*L, 512] (cell3)
  const int lane = threadIdx.x & 31;
  const int wave = threadIdx.x >> 5;           // 0..3 -> M tile
  const int m0   = wave * 16;
  const int n0   = blockIdx.x * 16;
  const int pos  = lane & 15;
  const int khb  = (lane >> 4) * 2;            // K pair {0,1} or {2,3}
  const bool pfl = (lane >> 4) == 0;           // one prefetch lane-set per row

  v8f ri = {}, zi = {}, ni = {}, rh = {}, zh = {}, nh = {};

  if (!ONE_HOT) {
    const float* arow = X + (size_t)(m0 + pos) * IN_DIM + khb;
    const float* br  = Wih + (size_t)(n0 + pos) * IN_DIM + khb;
    const float* bz  = br + (size_t)H_DIM * IN_DIM;
    const float* bn_ = bz + (size_t)H_DIM * IN_DIM;
    for (int k0 = 0; k0 < IN_DIM; k0 += 64) {   // IN_DIM % 64 == 0 (768/512)
      if (pfl) {                                 // speculative prefetch +256B
        __builtin_prefetch(arow + k0 + 64, 0, 3);
        __builtin_prefetch(br  + k0 + 64, 0, 3);
        __builtin_prefetch(bz  + k0 + 64, 0, 3);
        __builtin_prefetch(bn_ + k0 + 64, 0, 3);
      }
      #pragma unroll
      for (int k1 = 0; k1 < 64; k1 += 4) {
        int k = k0 + k1;
        v2f a = *(const v2f*)(arow + k);
        ri = wmma_k4(a, *(const v2f*)(br  + k), ri);
        zi = wmma_k4(a, *(const v2f*)(bz  + k), zi);
        ni = wmma_k4(a, *(const v2f*)(bn_ + k), ni);
      }
    }
  }
  {
    const float* arow = Hprev + (size_t)(m0 + pos) * H_DIM + khb;
    const float* br  = Whh + (size_t)(n0 + pos) * H_DIM + khb;
    const float* bz  = br + (size_t)H_DIM * H_DIM;
    const float* bn_ = bz + (size_t)H_DIM * H_DIM;
    for (int k0 = 0; k0 < H_DIM; k0 += 64) {     // H_DIM % 64 == 0 (768/512)
      if (pfl) {
        __builtin_prefetch(arow + k0 + 64, 0, 3);
        __builtin_prefetch(br  + k0 + 64, 0, 3);
        __builtin_prefetch(bz  + k0 + 64, 0, 3);
        __builtin_prefetch(bn_ + k0 + 64, 0, 3);
      }
      #pragma unroll
      for (int k1 = 0; k1 < 64; k1 += 4) {
        int k = k0 + k1;
        v2f a = *(const v2f*)(arow + k);
        rh = wmma_k4(a, *(const v2f*)(br  + k), rh);
        zh = wmma_k4(a, *(const v2f*)(bz  + k), zh);
        nh = wmma_k4(a, *(const v2f*)(bn_ + k), nh);
      }
    }
  }

  const int hi8 = (lane >> 4) * 8;
  const int col = n0 + (lane & 15);
  const float b_ri = bih[col], b_zi = bih[H_DIM + col], b_ni = bih[2 * H_DIM + col];
  const float b_rh = bhh[col], b_zh = bhh[H_DIM + col], b_nh = bhh[2 * H_DIM + col];

  #pragma unroll
  for (int e = 0; e < 8; ++e) {
    int row = m0 + e + hi8;                    // batch index
    float gir, giz, gin;
    if (ONE_HOT) {
      int c = seq[row * SEQ_L + t];
      gir = Wih[(size_t)col * IN_DIM + c] + b_ri;
      giz = Wih[(size_t)(H_DIM + col) * IN_DIM + c] + b_zi;
      gin = Wih[(size_t)(2 * H_DIM + col) * IN_DIM + c] + b_ni;
    } else {
      gir = ri[e] + b_ri;
      giz = zi[e] + b_zi;
      gin = ni[e] + b_ni;
    }
    float r = sigmoidf_(gir + rh[e] + b_rh);
    float z = sigmoidf_(giz + zh[e] + b_zh);
    float n = tanhf_(gin + r * (nh[e] + b_nh));
    float hp = Hprev[(size_t)row * H_DIM + col];
    float h = (1.0f - z) * n + z * hp;
    Hout[(size_t)row * H_DIM + col] = h;
    if (outs) outs[((size_t)row * SEQ_L + t) * HID + col] = h;   // batch-major flatten
  }
}

// ---------------------------------------------------------------------------
// BatchNorm stats over 16000 rows (biased variance, matching jnp var ddof=0).
// ---------------------------------------------------------------------------
__global__ void bn_stats(const float* __restrict__ X, int rows,
                         float* __restrict__ mean, float* __restrict__ var) {
  const int col = blockIdx.x;
  const int tid = threadIdx.x;
  __shared__ float rs[256], rq[256];
  float s = 0.f, q = 0.f;
  for (int r = tid; r < rows; r += 256) {
    float v = X[(size_t)r * HID + col];
    s += v;
    q += v * v;
  }
  rs[tid] = s; rq[tid] = q;
  __syncthreads();
  for (int off = 128; off > 0; off >>= 1) {
    if (tid < off) { rs[tid] += rs[tid + off]; rq[tid] += rq[tid + off]; }
    __syncthreads();
  }
  if (tid == 0) {
    float m = rs[0] / (float)rows;
    mean[col] = m;
    var[col]  = rq[0] / (float)rows - m * m;
  }
}

__global__ void bn_act(const float* __restrict__ X,
                       const float* __restrict__ mean, const float* __restrict__ var,
                       const float* __restrict__ g, const float* __restrict__ be,
                       float* __restrict__ Y, int n) {
  int i = blockIdx.x * 256 + threadIdx.x;
  if (i >= n) return;
  int col = i & (HID - 1);
  float xn = g[col] * (X[i] - mean[col]) * rsqrtf(var[col] + BN_EPS) + be[col];
  Y[i] = sigmoidf_(xn);
}

// ---------------------------------------------------------------------------
// Scoring GEMM: Y[M,NOUT] = A[M,K] @ W[NOUT,K]^T + bias.  Block = 8 waves over
// a 64x32 output region; OOB W rows clamped (final head N=34 padded to tiles).
// ---------------------------------------------------------------------------
template <int KDIM, int NOUT, int LDY>
__global__ void score_gemm(const float* __restrict__ A, const float* __restrict__ W,
                           const float* __restrict__ bias, float* __restrict__ Y) {
  const int lane = threadIdx.x & 31;
  const int wave = threadIdx.x >> 5;           // 0..7
  const int m0 = blockIdx.x * 64 + (wave >> 1) * 16;
  const int n0 = blockIdx.y * 32 + (wave & 1) * 16;
  const int pos = lane & 15;
  const int khb = (lane >> 4) * 2;
  const bool pfl = (lane >> 4) == 0;

  v8f acc = {};
  const float* arow = A + (size_t)(m0 + pos) * KDIM + khb;
  int wr = n0 + pos;
  if (wr >= NOUT) wr = 0;                      // clamp padded rows (stores guarded)
  const float* brow = W + (size_t)wr * KDIM + khb;
  for (int k0 = 0; k0 < KDIM; k0 += 64) {      // KDIM % 64 == 0 (512)
    if (pfl) {
      __builtin_prefetch(arow + k0 + 64, 0, 3);
      __builtin_prefetch(brow + k0 + 64, 0, 3);
    }
    #pragma unroll
    for (int k1 = 0; k1 < 64; k1 += 4) {
      int k = k0 + k1;
      acc = wmma_k4(*(const v2f*)(arow + k), *(const v2f*)(brow + k), acc);
    }
  }

  const int hi8 = (lane >> 4) * 8;
  const int col = n0 + (lane & 15);
  if (col < NOUT) {
    float bv = bias[col];
    #pragma unroll
    for (int e = 0; e < 8; ++e) {
      int row = m0 + e + hi8;
      Y[(size_t)row * LDY + col] = acc[e] + bv;
    }
  }
}

// ---------------------------------------------------------------------------
// Host launcher.  d_in order = setup_inputs() dict order with params flattened
// in insertion order (first_hidden, init_h1..3, cell0..3{w_ih,w_hh,b_ih,b_hh},
// wq,bq,g1,be1,w1,bb1,g2,be2,w2,bb2).
// ---------------------------------------------------------------------------
extern "C" void kernel_launch(void* const* d_in, const int* in_sizes, int n_in,
                              void* d_out, int out_size, void* d_ws, size_t ws_size,
                              hipStream_t stream) {
  const int*   seq    = (const int*)  d_in[0];
  const float* keys   = (const float*)d_in[1];
  const float* values = (const float*)d_in[2];
  const int*   masks  = (const int*)  d_in[3];
  const float* fh     = (const float*)d_in[4];
  const float* ih1    = (const float*)d_in[5];
  const float* ih2    = (const float*)d_in[6];
  const float* ih3    = (const float*)d_in[7];
  const float* w_ih0  = (const float*)d_in[8];
  const float* w_hh0  = (const float*)d_in[9];
  const float* b_ih0  = (const float*)d_in[10];
  const float* b_hh0  = (const float*)d_in[11];
  const float* w_ih1  = (const float*)d_in[12];
  const float* w_hh1  = (const float*)d_in[13];
  const float* b_ih1  = (const float*)d_in[14];
  const float* b_hh1  = (const float*)d_in[15];
  const float* w_ih2  = (const float*)d_in[16];
  const float* w_hh2  = (const float*)d_in[17];
  const float* b_ih2  = (const float*)d_in[18];
  const float* b_hh2  = (const float*)d_in[19];
  const float* w_ih3  = (const float*)d_in[20];
  const float* w_hh3  = (const float*)d_in[21];
  const float* b_ih3  = (const float*)d_in[22];
  const float* b_hh3  = (const float*)d_in[23];
  const float* wq     = (const float*)d_in[24];
  const float* bq     = (const float*)d_in[25];
  const float* g1     = (const float*)d_in[26];
  const float* be1    = (const float*)d_in[27];
  const float* w1     = (const float*)d_in[28];
  const float* bb1    = (const float*)d_in[29];
  const float* g2     = (const float*)d_in[30];
  const float* be2    = (const float*)d_in[31];
  const float* w2     = (const float*)d_in[32];
  const float* bb2    = (const float*)d_in[33];
  (void)in_sizes; (void)n_in; (void)out_size; (void)ws_size;

  // ---- workspace carve-out (~67 MB fp32, all fully written every launch) ----
  float* ws = (float*)d_ws;
  size_t off = 0;
  auto carve = [&](size_t n) { float* p = ws + off; off += (n + 63) & ~(size_t)63; return p; };
  float* h0init = carve((size_t)BATCH * G0);          // [64,768]  (ctx | fh)
  float* h0buf  = carve((size_t)BATCH * G0);          // [64,768]
  float* h1buf  = carve((size_t)2 * BATCH * HID);     // ping-pong
  float* h2buf  = carve((size_t)2 * BATCH * HID);
  float* h3buf  = carve((size_t)2 * BATCH * HID);
  float* outs   = carve((size_t)BATCH * SEQ_L * HID); // [16000,512] h3 trace / reused as y1
  float* abuf   = carve((size_t)BATCH * SEQ_L * HID); // activations
  float* mean1  = carve(HID);
  float* var1   = carve(HID);
  float* mean2  = carve(HID);
  float* var2   = carve(HID);

  const size_t BH = (size_t)BATCH * HID;

  init_kernel<<<(BATCH * HID) / 256, 256, 0, stream>>>(fh, ih1, ih2, ih3,
                                                       h0init, h1buf, h2buf, h3buf);

  for (int t = 0; t < SEQ_L; ++t) {
    const int p = t & 1;
    const float* h1i = h1buf + (size_t)p * BH;  float* h1o = h1buf + (size_t)(1 - p) * BH;
    const float* h2i = h2buf + (size_t)p * BH;  float* h2o = h2buf + (size_t)(1 - p) * BH;
    const float* h3i = h3buf + (size_t)p * BH;  float* h3o = h3buf + (size_t)(1 - p) * BH;

    attn_kernel<<<BATCH, 256, 0, stream>>>(h3i, wq, bq, keys, values, masks, h0init);
    gru_kernel<NUM_CHARS, G0, true ><<<G0 / 16, 128, 0, stream>>>(
        nullptr, h0init, w_ih0, w_hh0, b_ih0, b_hh0, seq, t, h0buf, nullptr);
    gru_kernel<G0, HID, false><<<HID / 16, 128, 0, stream>>>(
        h0buf, h1i, w_ih1, w_hh1, b_ih1, b_hh1, nullptr, t, h1o, nullptr);
    gru_kernel<HID, HID, false><<<HID / 16, 128, 0, stream>>>(
        h1o, h2i, w_ih2, w_hh2, b_ih2, b_hh2, nullptr, t, h2o, nullptr);
    gru_kernel<HID, HID, false><<<HID / 16, 128, 0, stream>>>(
        h2o, h3i, w_ih3, w_hh3, b_ih3, b_hh3, nullptr, t, h3o, outs);
  }

  // ---- scoring head: BN -> sigmoid -> GEMM -> BN -> sigmoid -> GEMM ----
  const int ROWS = BATCH * SEQ_L;  // 16000
  bn_stats<<<HID, 256, 0, stream>>>(outs, ROWS, mean1, var1);
  bn_act<<<(ROWS * HID + 255) / 256, 256, 0, stream>>>(outs, mean1, var1, g1, be1,
                                                       abuf, ROWS * HID);
  score_gemm<HID, HID, HID><<<dim3(ROWS / 64, HID / 32), 256, 0, stream>>>(
      abuf, w1, bb1, outs);                    // y1 reuses `outs`
  bn_stats<<<HID, 256, 0, stream>>>(outs, ROWS, mean2, var2);
  bn_act<<<(ROWS * HID + 255) / 256, 256, 0, stream>>>(outs, mean2, var2, g2, be2,
                                                       abuf, ROWS * HID);
  score_gemm<HID, NUM_CHARS, NUM_CHARS><<<dim3(ROWS / 64, 2), 256, 0, stream>>>(
      abuf, w2, bb2, (float*)d_out);
}